// ExpansionInfoModule_60181081751958
// MI455X (gfx1250) — compile-verified
//
#include <hip/hip_runtime.h>
#include <math.h>

// ---------------------------------------------------------------------------
// MI455X (gfx1250) implementation.
// All 1x1 convs + the 3x3/s2 pool conv run as fp32 WMMA GEMMs
// (V_WMMA_F32_16X16X4_F32, wave32). Each wave computes a 16x64 D tile
// (4 accumulators) with a preloaded A panel and immediate-offset B loads.
// The 8 fixed "edge" depthwise kernels are algebraically eliminated
// (out = v[n+off] - v[n]) so the 8x expanded key/value tensors are never
// materialized -- scores and aggregation read the base maps with shifts.
// ---------------------------------------------------------------------------

typedef __attribute__((ext_vector_type(2))) float v2f;
typedef __attribute__((ext_vector_type(8))) float v8f;

__device__ __forceinline__ v8f wmma_f32_16x16x4(v2f a, v2f b, v8f c) {
  // 8 args: (neg_a, A, neg_b, B, c_mod, C, reuse_a, reuse_b)
  return __builtin_amdgcn_wmma_f32_16x16x4_f32(false, a, false, b, (short)0, c,
                                               false, false);
}

// 8 edge-kernel neighbor offsets (dy,dx), e=0..3 from d1, e=4..7 mirrored (d2)
__constant__ int c_offy[8] = {-1, -1, -1, 0, 1, 1, 1, 0};
__constant__ int c_offx[8] = {-1, 0, 1, 1, 1, 0, -1, -1};

// ---------------------------------------------------------------------------
// Generic 1x1-conv GEMM:  Y[b][o][n] = sum_i W[o][i] * X[b][i][n]
// CIN, N compile-time -> fully unrolled K loop, immediate-offset loads.
// grid: (N/256, Cout/16, B), block 128 (4 waves, each one 16x64 D tile).
// ---------------------------------------------------------------------------
template <int CIN, int N>
__global__ __launch_bounds__(128) void gemm1x1_wmma(
    const float* __restrict__ W, const float* __restrict__ X,
    float* __restrict__ Y, int Cout) {
  const int lane = threadIdx.x & 31;
  const int wave = threadIdx.x >> 5;
  const int n0 = (blockIdx.x * 4 + wave) * 64;
  const int o0 = blockIdx.y * 16;
  const int b = blockIdx.z;
  const float* Xb = X + (size_t)b * CIN * N;
  float* Yb = Y + (size_t)b * Cout * N;

  const int m = lane & 15;       // A row / B,D column within 16-wide tile
  const int half = lane >> 4;    // 0: K={0,1} rows 0-7 ; 1: K={2,3} rows 8-15
  const int khalf = half * 2;

  // Preload full A panel for this M tile (CIN/4 v2f fragments).
  v2f a[CIN / 4];
  const float* Wp = W + (o0 + m) * CIN + khalf;
#pragma unroll
  for (int k = 0; k < CIN / 4; ++k) {
    a[k].x = Wp[4 * k];
    a[k].y = Wp[4 * k + 1];
  }

  v8f acc[4];
#pragma unroll
  for (int t = 0; t < 4; ++t)
    acc[t] = (v8f){0.f, 0.f, 0.f, 0.f, 0.f, 0.f, 0.f, 0.f};

  const float* p = Xb + (size_t)khalf * N + n0 + m;
#pragma unroll
  for (int k = 0; k < CIN / 4; ++k) {
#pragma unroll
    for (int t = 0; t < 4; ++t) {
      v2f bb;
      bb.x = p[t * 16];      // row k*4+khalf,   immediate offset
      bb.y = p[N + t * 16];  // row k*4+khalf+1, immediate offset (4N <= 256KB)
      acc[t] = wmma_f32_16x16x4(a[k], bb, acc[t]);
    }
    p += 4 * (size_t)N;  // one pointer bump per k-step
  }

  float* Ys = Yb + (size_t)(o0 + half * 8) * N + n0 + m;
#pragma unroll
  for (int j = 0; j < 8; ++j)
#pragma unroll
    for (int t = 0; t < 4; ++t) Ys[(size_t)j * N + t * 16] = acc[t][j];
}

// ---------------------------------------------------------------------------
// pool conv: 3x3, stride 2, pad 1.  cen (B,64,256,256) -> pooled (B,64,128,128)
// 9 accumulated WMMA GEMM taps; zero-pad via clamped-address load + select.
// grid: (16384/256, 4, B), block 128 (4 waves x 16x64 tile).
// ---------------------------------------------------------------------------
__global__ __launch_bounds__(128) void pool_conv_wmma(
    const float* __restrict__ Wp, const float* __restrict__ cen,
    float* __restrict__ pooled) {
  const int lane = threadIdx.x & 31;
  const int wave = threadIdx.x >> 5;
  const int n0 = (blockIdx.x * 4 + wave) * 64;
  const int o0 = blockIdx.y * 16;
  const int b = blockIdx.z;
  const int m = lane & 15;
  const int half = lane >> 4;
  const int khalf = half * 2;
  const float* cb = cen + (size_t)b * 64 * 65536 + (size_t)khalf * 65536;

  v8f acc[4];
#pragma unroll
  for (int t = 0; t < 4; ++t)
    acc[t] = (v8f){0.f, 0.f, 0.f, 0.f, 0.f, 0.f, 0.f, 0.f};

  for (int rc = 0; rc < 9; ++rc) {
    const int r = rc / 3, c = rc % 3;
    // A base for this tap: stride per k-step is 4*9 floats (immediate ok).
    const float* Wrc = Wp + ((size_t)(o0 + m) * 64 + khalf) * 9 + rc;
    // Per-subtile gather pixel (clamped) + validity.
    int ip[4];
    float msk[4];
#pragma unroll
    for (int t = 0; t < 4; ++t) {
      const int pix = n0 + t * 16 + m;
      const int iy = 2 * (pix >> 7) + r - 1;
      const int ix = 2 * (pix & 127) + c - 1;
      const bool inb = (iy >= 0) & (iy < 256) & (ix >= 0) & (ix < 256);
      ip[t] = inb ? (iy * 256 + ix) : 0;  // clamp: always a valid address
      msk[t] = inb ? 1.f : 0.f;
    }
    const float* pk = cb;
#pragma unroll
    for (int k = 0; k < 16; ++k) {
      v2f a;
      a.x = Wrc[36 * k];      // (k*4+khalf, r, c)
      a.y = Wrc[36 * k + 9];  // (k*4+khalf+1, r, c)
#pragma unroll
      for (int t = 0; t < 4; ++t) {
        v2f bb;
        bb.x = pk[ip[t]] * msk[t];
        bb.y = pk[65536 + ip[t]] * msk[t];
        acc[t] = wmma_f32_16x16x4(a, bb, acc[t]);
      }
      pk += 4 * 65536;
    }
  }

  float* pb =
      pooled + (size_t)b * 64 * 16384 + (size_t)(o0 + half * 8) * 16384 + n0 + m;
#pragma unroll
  for (int j = 0; j < 8; ++j)
#pragma unroll
    for (int t = 0; t < 4; ++t) pb[(size_t)j * 16384 + t * 16] = acc[t][j];
}

// ---------------------------------------------------------------------------
// Scores: for each (b,c,k) compute dot(q_raw, ke) and ||ke||^2 over size=16384.
// k=0: ke = k0.  k=1+8i+e: ke[s] = pad(ks_i[s+off_e*d_i]) - ks_i[s].
// grid: (B*C, 25), block 256.
// ---------------------------------------------------------------------------
__global__ __launch_bounds__(256) void score_dot_kernel(
    const float* __restrict__ q, const float* __restrict__ k0,
    const float* __restrict__ ks, float* __restrict__ dots,
    float* __restrict__ nrms) {
  const int bc = blockIdx.x;  // 0..127
  const int k = blockIdx.y;   // 0..24
  const int tid = threadIdx.x;
  const float* qv = q + (size_t)bc * 16384;
  float dot = 0.f, nrm = 0.f;
  if (k == 0) {
    const float* kv = k0 + (size_t)bc * 16384;
    for (int s = tid; s < 16384; s += 256) {
      float ke = kv[s];
      dot += qv[s] * ke;
      nrm += ke * ke;
    }
  } else {
    const int i = (k - 1) >> 3, e = (k - 1) & 7;
    const int d = 1 << i;  // SHIFTS = 1,2,4
    const int dy = c_offy[e] * d, dx = c_offx[e] * d;
    const float* kv = ks + (size_t)i * (128 * 16384) + (size_t)bc * 16384;
    for (int s = tid; s < 16384; s += 256) {
      const int sy = s >> 7, sx = s & 127;
      const int ny = sy + dy, nx = sx + dx;
      const bool inb = (ny >= 0) & (ny < 128) & (nx >= 0) & (nx < 128);
      const int na = inb ? (ny * 128 + nx) : 0;  // clamped address
      float ke = (inb ? kv[na] : 0.f) - kv[s];
      dot += qv[s] * ke;
      nrm += ke * ke;
    }
  }
  __shared__ float sd[256], sn[256];
  sd[tid] = dot;
  sn[tid] = nrm;
  __syncthreads();
  for (int off = 128; off > 0; off >>= 1) {
    if (tid < off) {
      sd[tid] += sd[tid + off];
      sn[tid] += sn[tid + off];
    }
    __syncthreads();
  }
  if (tid == 0) {
    dots[bc * 25 + k] = sd[0];
    nrms[bc * 25 + k] = sn[0];
  }
}

// Normalize keys (divide by ||ke||), then l2-normalize score over K=25.
// (Normalizing q is a uniform scale per (b,c) and cancels here, so skipped.)
__global__ void score_norm_kernel(const float* __restrict__ dots,
                                  const float* __restrict__ nrms,
                                  float* __restrict__ score) {
  const int bc = threadIdx.x;
  if (bc >= 128) return;
  float s[25];
  float ss = 0.f;
#pragma unroll
  for (int k = 0; k < 25; ++k) {
    float kn = sqrtf(nrms[bc * 25 + k]);
    float v = dots[bc * 25 + k] / fmaxf(kn, 1e-12f);
    s[k] = v;
    ss += v * v;
  }
  const float inv = 1.f / fmaxf(sqrtf(ss), 1e-12f);
#pragma unroll
  for (int k = 0; k < 25; ++k) score[bc * 25 + k] = s[k] * inv;
}

// ---------------------------------------------------------------------------
// Aggregation: out[b,c,n] = s0*v0[n] + sum_i [ sum_e s_ie*pad(vs_i[n+off])
//                                              - (sum_e s_ie)*vs_i[n] ]
// grid: (hw/256, C, B), block 256.
// ---------------------------------------------------------------------------
__global__ __launch_bounds__(256) void aggregate_kernel(
    const float* __restrict__ score, const float* __restrict__ v0,
    const float* __restrict__ vs, float* __restrict__ out) {
  const int n = blockIdx.x * 256 + threadIdx.x;
  const int c = blockIdx.y, b = blockIdx.z;
  const int bc = b * 64 + c;
  __shared__ float s[25];
  if (threadIdx.x < 25) s[threadIdx.x] = score[bc * 25 + threadIdx.x];
  __syncthreads();

  const int y = n >> 8, x = n & 255;
  float acc = s[0] * v0[(size_t)bc * 65536 + n];
#pragma unroll
  for (int i = 0; i < 3; ++i) {
    const float* vv = vs + (size_t)i * (128 * 65536) + (size_t)bc * 65536;
    const int d = 1 << i;
    const float center = vv[n];
    float a = 0.f, csum = 0.f;
#pragma unroll
    for (int e = 0; e < 8; ++e) {
      const float sv = s[1 + i * 8 + e];
      csum += sv;
      const int ny = y + c_offy[e] * d, nx = x + c_offx[e] * d;
      const bool inb = (ny >= 0) & (ny < 256) & (nx >= 0) & (nx < 256);
      const int na = inb ? (ny * 256 + nx) : 0;
      a += inb ? (sv * vv[na]) : 0.f;
    }
    acc += a - csum * center;
  }
  out[(size_t)bc * 65536 + n] = acc;
}

// ---------------------------------------------------------------------------
// BatchNorm (training-mode batch stats) reductions and apply+ReLU.
// ---------------------------------------------------------------------------
__global__ __launch_bounds__(256) void bn_stats_kernel(
    const float* __restrict__ X, float* __restrict__ mean,
    float* __restrict__ var, int C, int N) {
  const int c = blockIdx.x, tid = threadIdx.x;
  float s = 0.f, s2 = 0.f;
  for (int b = 0; b < 2; ++b) {
    const float* xb = X + ((size_t)b * C + c) * N;
    for (int n = tid; n < N; n += 256) {
      float v = xb[n];
      s += v;
      s2 += v * v;
    }
  }
  __shared__ float sa[256], sb[256];
  sa[tid] = s;
  sb[tid] = s2;
  __syncthreads();
  for (int off = 128; off > 0; off >>= 1) {
    if (tid < off) {
      sa[tid] += sa[tid + off];
      sb[tid] += sb[tid + off];
    }
    __syncthreads();
  }
  if (tid == 0) {
    const float inv = 1.f / (2.f * (float)N);
    const float m = sa[0] * inv;
    mean[c] = m;
    var[c] = sb[0] * inv - m * m;  // biased var (jnp.var default)
  }
}

// float4-vectorized apply: 4 elements/thread (N % 4 == 0 so one channel/vec).
__global__ __launch_bounds__(256) void bn_relu_kernel(
    const float* __restrict__ X, const float* __restrict__ mean,
    const float* __restrict__ var, const float* __restrict__ gamma,
    const float* __restrict__ beta, float* __restrict__ Y, int C, int N) {
  const size_t i4 = (size_t)blockIdx.x * 256 + threadIdx.x;
  if (i4 * 4 >= (size_t)2 * C * N) return;
  const int c = (int)((i4 * 4 / (size_t)N) % (size_t)C);
  const float sc = gamma[c] * rsqrtf(var[c] + 1e-5f);
  const float sh = beta[c] - sc * mean[c];
  const float4 v = ((const float4*)X)[i4];
  float4 r;
  r.x = fmaxf(sc * v.x + sh, 0.f);
  r.y = fmaxf(sc * v.y + sh, 0.f);
  r.z = fmaxf(sc * v.z + sh, 0.f);
  r.w = fmaxf(sc * v.w + sh, 0.f);
  ((float4*)Y)[i4] = r;
}

// ---------------------------------------------------------------------------
// Host launcher
// ---------------------------------------------------------------------------
extern "C" void kernel_launch(void* const* d_in, const int* in_sizes, int n_in,
                              void* d_out, int out_size, void* d_ws,
                              size_t ws_size, hipStream_t stream) {
  const float* cen      = (const float*)d_in[0];   // (2,64,256,256)
  const float* pool_w   = (const float*)d_in[1];   // (64,64,3,3)
  const float* query_w  = (const float*)d_in[2];   // (64,64)
  const float* key_w    = (const float*)d_in[3];   // (64,64)
  const float* value_w  = (const float*)d_in[4];   // (64,64)
  const float* keysh_w  = (const float*)d_in[5];   // (3,64,64)
  const float* valsh_w  = (const float*)d_in[6];   // (3,64,64)
  // d_in[7] = edge kernels: fixed +/-1 pattern, folded analytically.
  const float* conv1_w  = (const float*)d_in[8];   // (128,64)
  const float* bn1_g    = (const float*)d_in[9];
  const float* bn1_b    = (const float*)d_in[10];
  const float* conv2_w  = (const float*)d_in[11];  // (64,128)
  const float* bn2_g    = (const float*)d_in[12];
  const float* bn2_b    = (const float*)d_in[13];
  float* outp = (float*)d_out;

  const int B = 2, C = 64, HW = 65536, SZ = 16384;
  float* ws = (float*)d_ws;
  // Workspace layout (floats); peak ~46.2M floats (~185 MB) with reuse.
  float* pooled = ws;                    // 2*64*16384
  float* q_raw  = pooled + 2 * C * SZ;   // 2*64*16384
  float* k0     = q_raw + 2 * C * SZ;    // 2*64*16384
  float* ks     = k0 + 2 * C * SZ;       // 3 * 2*64*16384
  float* v0     = ks + 3 * 2 * C * SZ;   // 2*64*65536
  float* vs     = v0 + 2 * C * HW;       // 3 * 2*64*65536
  float* dots   = vs + 3 * 2 * C * HW;   // 3200
  float* nrms   = dots + 3200;           // 3200
  float* score  = nrms + 3200;           // 3200
  float* m1     = score + 3200;          // 128
  float* var1   = m1 + 128;              // 128
  float* m2     = var1 + 128;            // 64
  float* var2   = m2 + 64;               // 64
  // Reused regions (producers of originals are finished before overwrite):
  float* attn_out = q_raw;  // 8.39M floats over q_raw+k0 (10.5M available)
  float* y1 = v0;           // 16.78M floats over v0/vs (33.5M available)
  float* y2 = v0 + 2 * 2 * C * HW;  // 8.39M floats, still inside v0+vs

  // 1) pooled = conv3x3 stride2 (WMMA)
  pool_conv_wmma<<<dim3(SZ / 256, C / 16, B), 128, 0, stream>>>(pool_w, cen,
                                                                pooled);
  // 2) 1x1 GEMMs (WMMA)
  gemm1x1_wmma<64, 16384><<<dim3(SZ / 256, C / 16, B), 128, 0, stream>>>(
      query_w, pooled, q_raw, C);
  gemm1x1_wmma<64, 16384><<<dim3(SZ / 256, C / 16, B), 128, 0, stream>>>(
      key_w, pooled, k0, C);
  for (int i = 0; i < 3; ++i)
    gemm1x1_wmma<64, 16384><<<dim3(SZ / 256, C / 16, B), 128, 0, stream>>>(
        keysh_w + i * C * C, pooled, ks + (size_t)i * 2 * C * SZ, C);
  gemm1x1_wmma<64, 65536><<<dim3(HW / 256, C / 16, B), 128, 0, stream>>>(
      value_w, cen, v0, C);
  for (int i = 0; i < 3; ++i)
    gemm1x1_wmma<64, 65536><<<dim3(HW / 256, C / 16, B), 128, 0, stream>>>(
        valsh_w + i * C * C, cen, vs + (size_t)i * 2 * C * HW, C);

  // 3) attention scores (edge kernels folded into shifted reads)
  score_dot_kernel<<<dim3(B * C, 25), 256, 0, stream>>>(q_raw, k0, ks, dots,
                                                        nrms);
  score_norm_kernel<<<1, 128, 0, stream>>>(dots, nrms, score);

  // 4) aggregation (writes over q_raw/k0 region; those are dead now)
  aggregate_kernel<<<dim3(HW / 256, C, B), 256, 0, stream>>>(score, v0, vs,
                                                             attn_out);

  // 5) conv1 -> BN -> ReLU (in place on y1)
  gemm1x1_wmma<64, 65536><<<dim3(HW / 256, (2 * C) / 16, B), 128, 0, stream>>>(
      conv1_w, attn_out, y1, 2 * C);
  bn_stats_kernel<<<2 * C, 256, 0, stream>>>(y1, m1, var1, 2 * C, HW);
  bn_relu_kernel<<<(2 * 2 * C * HW) / 1024, 256, 0, stream>>>(
      y1, m1, var1, bn1_g, bn1_b, y1, 2 * C, HW);
  // 6) conv2 -> BN -> ReLU -> d_out
  gemm1x1_wmma<128, 65536><<<dim3(HW / 256, C / 16, B), 128, 0, stream>>>(
      conv2_w, y1, y2, C);
  bn_stats_kernel<<<C, 256, 0, stream>>>(y2, m2, var2, C, HW);
  bn_relu_kernel<<<(2 * C * HW) / 1024, 256, 0, stream>>>(y2, m2, var2, bn2_g,
                                                          bn2_b, outp, C, HW);
  (void)in_sizes; (void)n_in; (void)out_size; (void)ws_size;
}